// InstanSeg_Torchscript_76802605187431
// MI455X (gfx1250) — compile-verified
//
#include <hip/hip_runtime.h>
#include <stdint.h>

// Problem constants (from reference: E=6, S=2, C=2000, ws=32 -> Hw=64, h=w=1024)
#define HWIDTH   1024
#define PLANE    (HWIDTH * HWIDTH)          // 1,048,576 elements per image plane
#define NROWS    8192000                     // C * Hw * Hw = 2000 * 4096
#define FEAT_ELEMS (NROWS * 8)               // 65,536,000 floats

// Native clang vector (HIP's float4 is a class and is rejected by the
// nontemporal builtins); lowers to one global_store_b128 th:TH_STORE_NT.
typedef float v4f __attribute__((ext_vector_type(4)));

// ---------------------------------------------------------------------------
// Kernel 1: feature gather.
// out[r, 0:6] = xe[addr] - c[ci, e];  out[r, 6:8] = sg{0,1}[addr]
// addr = mgf0[r]*1024 + mgf1[r], ci = r >> 12 (uniform per 256-thread block).
// Each plane base is a separate SGPR pair -> every gather is one
// global_load_b32 v, v_addr, s[base] scale_offset (no VALU address chains).
// One row (32 B) per lane -> wave32 writes 1 KB contiguous via 2x b128 NT stores.
// ---------------------------------------------------------------------------
__global__ __launch_bounds__(256) void feat_kernel(
    const float* __restrict__ x0, const float* __restrict__ x1,
    const float* __restrict__ x2, const float* __restrict__ x3,
    const float* __restrict__ x4, const float* __restrict__ x5,
    const float* __restrict__ sg0, const float* __restrict__ sg1,
    const float* __restrict__ c,      // (2000, 6)
    const int*   __restrict__ mgf0,
    const int*   __restrict__ mgf1,
    float*       __restrict__ out)    // (NROWS, 8)
{
    const int r = blockIdx.x * 256 + threadIdx.x;

    // ci is uniform across the block (4096 % 256 == 0) -> scalar loads for c row
    const int ci = (int)(blockIdx.x >> 4);
    const float* crow = c + ci * 6;
    const float c0 = crow[0], c1 = crow[1], c2 = crow[2];
    const float c3 = crow[3], c4 = crow[4], c5 = crow[5];

    // Prefetch the index stream well ahead (speculative, gfx1250 global_prefetch_b8)
    __builtin_prefetch(&mgf0[r + 16384], 0, 3);
    __builtin_prefetch(&mgf1[r + 16384], 0, 3);

    // Streamed-once index reads: non-temporal
    const int row = __builtin_nontemporal_load(&mgf0[r]);
    const int col = __builtin_nontemporal_load(&mgf1[r]);
    const int addr = row * HWIDTH + col;

    // Gathers: RT caching; 32 MB working set lives in the 192 MB L2.
    // All 8 issue back-to-back as saddr+scaled-vgpr loads.
    v4f lo, hi;
    lo.x = x0[addr] - c0;
    lo.y = x1[addr] - c1;
    lo.z = x2[addr] - c2;
    lo.w = x3[addr] - c3;
    hi.x = x4[addr] - c4;
    hi.y = x5[addr] - c5;
    hi.z = sg0[addr];
    hi.w = sg1[addr];

    // Write-once 262 MB output: non-temporal b128 stores
    v4f* dst = (v4f*)(out + (size_t)r * 8);
    __builtin_nontemporal_store(lo, dst + 0);
    __builtin_nontemporal_store(hi, dst + 1);
}

// ---------------------------------------------------------------------------
// Kernel 2: zero the per-pixel winner keys (8 MB in workspace).
// ---------------------------------------------------------------------------
__global__ __launch_bounds__(256) void zero_kernel(unsigned long long* __restrict__ win)
{
    const int i = blockIdx.x * 256 + threadIdx.x;
    win[i] = 0ull;
}

// ---------------------------------------------------------------------------
// Kernel 3: per-pixel argmax via u64 atomicMax.
// key = (prob_bits << 32) | (~index): higher prob wins; tie -> smaller index
// (matches stable lexsort((-prob, k)) + first-unique selection).
// ---------------------------------------------------------------------------
__global__ __launch_bounds__(256) void scatter_kernel(
    const float* __restrict__ prob,
    const int*   __restrict__ xc,
    const int*   __restrict__ yc,
    unsigned long long* __restrict__ win)
{
    const int i = blockIdx.x * 256 + threadIdx.x;
    const float p  = __builtin_nontemporal_load(&prob[i]);
    const int   cx = __builtin_nontemporal_load(&xc[i]);
    const int   cy = __builtin_nontemporal_load(&yc[i]);
    if (p >= 0.5f) {
        const int k = cx * HWIDTH + cy;                  // coords in [0,1024) -> always in range
        const unsigned int pb = __float_as_uint(p);      // p >= 0.5 > 0: bit pattern order-preserving, key != 0
        const unsigned long long key =
            ((unsigned long long)pb << 32) | (unsigned long long)(0xFFFFFFFFu - (unsigned int)i);
        atomicMax(&win[k], key);                          // global_atomic_max_u64 (L2-resident 8 MB)
    }
}

// ---------------------------------------------------------------------------
// Kernel 4: decode winners -> float label map. label = index/4096 + 1, else 0.
// ---------------------------------------------------------------------------
__global__ __launch_bounds__(256) void decode_kernel(
    const unsigned long long* __restrict__ win,
    float* __restrict__ label)
{
    const int k = blockIdx.x * 256 + threadIdx.x;
    const unsigned long long v = win[k];
    float lab = 0.0f;
    if (v != 0ull) {
        const unsigned int idx = 0xFFFFFFFFu - (unsigned int)(v & 0xFFFFFFFFull);
        lab = (float)((idx >> 12) + 1u);                 // idx/4096 + 1 == instance label
    }
    __builtin_nontemporal_store(lab, &label[k]);
}

// ---------------------------------------------------------------------------
extern "C" void kernel_launch(void* const* d_in, const int* in_sizes, int n_in,
                              void* d_out, int out_size, void* d_ws, size_t ws_size,
                              hipStream_t stream)
{
    (void)in_sizes; (void)n_in; (void)out_size; (void)ws_size;

    const float* x      = (const float*)d_in[0];   // (6, 1024, 1024)
    const float* c      = (const float*)d_in[1];   // (2000, 6)
    const float* sigma  = (const float*)d_in[2];   // (2, 1024, 1024)
    const float* prob   = (const float*)d_in[3];   // (2000, 1, 64, 64) flat
    const int*   mgf    = (const int*)d_in[4];     // (2, 8192000)
    const int*   coords = (const int*)d_in[5];     // (2, 2000, 1, 64, 64) flat
    // d_in[6] = window_size scalar (32), baked into constants

    float* feat  = (float*)d_out;                  // 65,536,000 floats
    float* label = (float*)d_out + FEAT_ELEMS;     // 1,048,576 floats
    unsigned long long* win = (unsigned long long*)d_ws;  // 8 MB scratch

    // Part 1: feature gather (independent of part 2)
    feat_kernel<<<NROWS / 256, 256, 0, stream>>>(
        x + 0 * PLANE, x + 1 * PLANE, x + 2 * PLANE,
        x + 3 * PLANE, x + 4 * PLANE, x + 5 * PLANE,
        sigma + 0 * PLANE, sigma + 1 * PLANE,
        c, mgf, mgf + NROWS, feat);

    // Part 2: zero -> atomic argmax scatter -> decode (stream-ordered)
    zero_kernel   <<<PLANE / 256, 256, 0, stream>>>(win);
    scatter_kernel<<<NROWS / 256, 256, 0, stream>>>(prob, coords, coords + NROWS, win);
    decode_kernel <<<PLANE / 256, 256, 0, stream>>>(win, label);
}